// RadialProfile_53738630807967
// MI455X (gfx1250) — compile-verified
//
#include <hip/hip_runtime.h>
#include <math.h>
#include <stdint.h>

// MI455X (gfx1250) RadialProfile: DFT-as-WMMA-GEMM formulation.
// F = W * X * W (W = symmetric 512x512 DFT matrix) via two f32 WMMA GEMM
// passes; power spectrum + fftshift + radial histogram fused into pass 2.
// - Tile staging global->LDS via Tensor Data Mover (one descriptor per wave),
//   DOUBLE-BUFFERED: TDM for chunk kb+1 overlaps WMMA compute of chunk kb.
// - TDM hardware row-padding provides the LDS bank-conflict padding.
// - W is symmetric and T is stored transposed, so every LDS tile is loaded
//   K-contiguous: all WMMA fragments are single aligned ds_load_b64 ops.

typedef float v2f __attribute__((ext_vector_type(2)));
typedef float v8f __attribute__((ext_vector_type(8)));
typedef unsigned int u32x4 __attribute__((ext_vector_type(4)));
typedef int i32x4 __attribute__((ext_vector_type(4)));
typedef int i32x8 __attribute__((ext_vector_type(8)));

#define NN 512
#define IMG (NN * NN)          // 262144
#define NBINS 363              // MAX_R + 1
#define NIMG 96                // 32 batches * 3 channels
#define NB 255                 // output bins per batch (1..255)
#define TPAD 20                // LDS tile row stride: 16 data + 4 pad floats

static __device__ __forceinline__ v8f wmma_f32x4(v2f a, v2f b, v8f c) {
  // D = A(16x4,f32) * B(4x16,f32) + C(16x16,f32)
  return __builtin_amdgcn_wmma_f32_16x16x4_f32(false, a, false, b, (short)0, c,
                                               false, false);
}

static __device__ __forceinline__ unsigned lds_off(const void* p) {
  // LDS aperture: flat addr low 32 bits are the LDS byte address (ISA 10.2).
  return (unsigned)(uintptr_t)p;
}

// Loop-invariant part of the TDM D# (group 1): 16 floats/row, rows_d1 rows,
// global row stride 512 f32, LDS pad 4 DW after every 16 DW (-> stride TPAD).
static __device__ __forceinline__ i32x8 tdm_g1_init(unsigned rows_d1) {
  i32x8 g1;
  g1[0] = (int)((2u << 16) |                           // data_size = 4B
                (1u << 20) |                           // pad_enable
                (3u << 22) | (3u << 25));              // every 16 DW, pad 4 DW
  g1[1] = (int)((NN & 0xFFFFu) << 16);                 // tensor_dim0 lo16
  g1[2] = (int)((NN >> 16) | ((NN & 0xFFFFu) << 16));  // dim0 hi | dim1 lo
  g1[3] = (int)((NN >> 16) | (16u << 16));             // dim1 hi | tile_dim0=16
  g1[4] = (int)(rows_d1 & 0xFFFFu);                    // tile_dim1, tile_dim2=0
  g1[5] = (int)NN;                                     // tensor_dim0_stride
  g1[6] = 0;
  g1[7] = 0;
  return g1;
}

static __device__ __forceinline__ void tdm_issue(const i32x8& g1,
                                                 unsigned lds_addr,
                                                 const float* gsrc) {
  unsigned long long ga = (unsigned long long)(uintptr_t)gsrc;
  u32x4 g0;
  g0[0] = 1u;                                          // count=1, user mode
  g0[1] = lds_addr;
  g0[2] = (unsigned)(ga & 0xFFFFFFFFu);
  g0[3] = (unsigned)((ga >> 32) & 0x01FFFFFFu) | (2u << 30);  // hi | type=2
  i32x4 gz = {0, 0, 0, 0};
#if __clang_major__ >= 23
  i32x8 gz8 = {0, 0, 0, 0, 0, 0, 0, 0};
  __builtin_amdgcn_tensor_load_to_lds(g0, g1, gz, gz, gz8, 0);
#else
  __builtin_amdgcn_tensor_load_to_lds(g0, g1, gz, gz, 0);
#endif
}

// ---------------------------------------------------------------- W builder
__global__ void k_build_w(float* __restrict__ Wr, float* __restrict__ Wi) {
  int i = blockIdx.x * blockDim.x + threadIdx.x;  // 0..IMG-1
  int j = i >> 9;
  int k = i & 511;
  int m = (j * k) & 511;  // exact periodic reduction -> high-precision angle
  float ang = -6.283185307179586f * (float)m * (1.0f / 512.0f);
  float s, c;
  __sincosf(ang, &s, &c);
  Wr[i] = c;
  Wi[i] = s;
}

// ------------------------------------------------------- zero bins + counts
__global__ void k_zero(float* __restrict__ p, int n) {
  int i = blockIdx.x * blockDim.x + threadIdx.x;
  if (i < n) p[i] = 0.0f;
}

__global__ void k_count(const int* __restrict__ rbin, float* __restrict__ counts) {
  int i = blockIdx.x * blockDim.x + threadIdx.x;  // 0..IMG-1
  atomicAdd(&counts[rbin[i]], 1.0f);              // counts < 2^24 -> exact f32
}

// --------------------- Stage 1: T = X * W (row FFT), stored TRANSPOSED -----
// Tt[v][j] = T[j][v]; B-side W tiles loaded transposed via W symmetry.
__global__ __launch_bounds__(128) void k_fft_rows(
    const float* __restrict__ X, const float* __restrict__ Wr,
    const float* __restrict__ Wi, float* __restrict__ T, int g0) {
  __shared__ __align__(16) float sA[2][64][TPAD];    // X[m][k]      rows m
  __shared__ __align__(16) float sBr[2][64][TPAD];   // Wr[n][k](=T) rows n
  __shared__ __align__(16) float sBi[2][64][TPAD];   // Wi[n][k](=T) rows n

  const int tid = threadIdx.x;
  const int lane = tid & 31;
  const int wv = tid >> 5;
  const int l15 = lane & 15;
  const int koff = (lane >> 4) << 1;  // lanes 0-15 -> K{0,1}; 16-31 -> K{2,3}

  const int tn = blockIdx.x, tm = blockIdx.y, z = blockIdx.z;
  const int img = g0 + z;
  const float* __restrict__ Xb = X + (size_t)img * IMG;
  float* __restrict__ Trt = T + (size_t)z * 2 * IMG;  // transposed [v][j]
  float* __restrict__ Tit = Trt + IMG;
  const int m0 = tm * 64, n0 = tn * 64;

  // Per-wave TDM assignment: X tile split in halves (waves 0,3) + Wr^T + Wi^T.
  const float* gbase;
  unsigned ldsb[2];
  unsigned d1;
  if (wv == 0 || wv == 3) {
    const int rh = (wv == 0) ? 0 : 32;
    gbase = Xb + (size_t)(m0 + rh) * NN;
    ldsb[0] = lds_off(&sA[0][rh][0]);
    ldsb[1] = lds_off(&sA[1][rh][0]);
    d1 = 32;
  } else if (wv == 1) {
    gbase = Wr + (size_t)n0 * NN;  // W symmetric: rows n, cols k
    ldsb[0] = lds_off(&sBr[0][0][0]);
    ldsb[1] = lds_off(&sBr[1][0][0]);
    d1 = 64;
  } else {
    gbase = Wi + (size_t)n0 * NN;
    ldsb[0] = lds_off(&sBi[0][0][0]);
    ldsb[1] = lds_off(&sBi[1][0][0]);
    d1 = 64;
  }
  const i32x8 g1 = tdm_g1_init(d1);

  const v8f zv = {0.f, 0.f, 0.f, 0.f, 0.f, 0.f, 0.f, 0.f};
  v8f accR[4], accI[4];
#pragma unroll
  for (int s = 0; s < 4; ++s) { accR[s] = zv; accI[s] = zv; }

  // Prologue: stage chunk 0 into buffer 0.
  tdm_issue(g1, ldsb[0], gbase);
  __builtin_amdgcn_s_wait_tensorcnt(0);
  __syncthreads();

  for (int kb = 0; kb < 32; ++kb) {
    const int cur = kb & 1;
    // Prefetch next chunk into the other buffer; overlaps the WMMA burst.
    if (kb < 31) tdm_issue(g1, ldsb[cur ^ 1], gbase + (kb + 1) * 16);

#pragma unroll
    for (int kk = 0; kk < 16; kk += 4) {
      v2f a = *(const v2f*)&sA[cur][wv * 16 + l15][kk + koff];  // aligned b64
      v2f br[4], bi[4];
#pragma unroll
      for (int s = 0; s < 4; ++s) {
        br[s] = *(const v2f*)&sBr[cur][s * 16 + l15][kk + koff];
        bi[s] = *(const v2f*)&sBi[cur][s * 16 + l15][kk + koff];
      }
#pragma unroll
      for (int s = 0; s < 4; ++s) {
        accR[s] = wmma_f32x4(a, br[s], accR[s]);
        accI[s] = wmma_f32x4(a, bi[s], accI[s]);
      }
    }

    __builtin_amdgcn_s_wait_tensorcnt(0);  // this wave's prefetch done
    __syncthreads();                       // all tiles visible / buffer free
  }

  // Transposed store: lane's 8 accumulator rows are contiguous in Tt[col][.]
  const int rbase = m0 + wv * 16 + ((lane >> 4) << 3);  // 32B-aligned run
#pragma unroll
  for (int s = 0; s < 4; ++s) {
    const int col = n0 + s * 16 + l15;
    *(v8f*)(Trt + (size_t)col * NN + rbase) = accR[s];
    *(v8f*)(Tit + (size_t)col * NN + rbase) = accI[s];
  }
}

// --------- Stage 2: F = W * T (col FFT) + |F|^2 + fftshift + radial binning
// Fr = Wr*Tr - Wi*Ti kept as separate accumulators P,Q; Fi shares one.
__global__ __launch_bounds__(128) void k_fft_cols_bin(
    const float* __restrict__ Wr, const float* __restrict__ Wi,
    const float* __restrict__ T, const int* __restrict__ rbin,
    float* __restrict__ binsum, int g0) {
  __shared__ __align__(16) float sAr[2][64][TPAD];   // Wr[u][j]  rows u
  __shared__ __align__(16) float sAi[2][64][TPAD];   // Wi[u][j]
  __shared__ __align__(16) float sBr[2][64][TPAD];   // Tt_r[v][j] rows v
  __shared__ __align__(16) float sBi[2][64][TPAD];   // Tt_i[v][j]
  __shared__ float lbins[NBINS];

  const int tid = threadIdx.x;
  const int lane = tid & 31;
  const int wv = tid >> 5;
  const int l15 = lane & 15;
  const int koff = (lane >> 4) << 1;

  const int tv = blockIdx.x, tu = blockIdx.y, z = blockIdx.z;
  const int img = g0 + z;
  const int batch = img / 3;  // channel folded; /3 applied in finalize
  const float* __restrict__ Trt = T + (size_t)z * 2 * IMG;  // [v][j]
  const float* __restrict__ Tit = Trt + IMG;
  const int u0 = tu * 64, v0 = tv * 64;

  for (int i = tid; i < NBINS; i += 128) lbins[i] = 0.0f;

  // Per-wave TDM assignment: Wr, Wi (A-side), Tt_r, Tt_i (B-side).
  const float* gbase;
  unsigned ldsb[2];
  if (wv == 0) {
    gbase = Wr + (size_t)u0 * NN;
    ldsb[0] = lds_off(&sAr[0][0][0]);
    ldsb[1] = lds_off(&sAr[1][0][0]);
  } else if (wv == 1) {
    gbase = Wi + (size_t)u0 * NN;
    ldsb[0] = lds_off(&sAi[0][0][0]);
    ldsb[1] = lds_off(&sAi[1][0][0]);
  } else if (wv == 2) {
    gbase = Trt + (size_t)v0 * NN;  // already transposed
    ldsb[0] = lds_off(&sBr[0][0][0]);
    ldsb[1] = lds_off(&sBr[1][0][0]);
  } else {
    gbase = Tit + (size_t)v0 * NN;
    ldsb[0] = lds_off(&sBi[0][0][0]);
    ldsb[1] = lds_off(&sBi[1][0][0]);
  }
  const i32x8 g1 = tdm_g1_init(64);

  const v8f zv = {0.f, 0.f, 0.f, 0.f, 0.f, 0.f, 0.f, 0.f};
  v8f fP[4], fQ[4], fI[4];  // P=Wr*Tr, Q=Wi*Ti, fI=Wr*Ti+Wi*Tr
#pragma unroll
  for (int s = 0; s < 4; ++s) { fP[s] = zv; fQ[s] = zv; fI[s] = zv; }

  // Prologue: stage chunk 0 into buffer 0.
  tdm_issue(g1, ldsb[0], gbase);
  __builtin_amdgcn_s_wait_tensorcnt(0);
  __syncthreads();

  for (int kb = 0; kb < 32; ++kb) {
    const int cur = kb & 1;
    if (kb < 31) tdm_issue(g1, ldsb[cur ^ 1], gbase + (kb + 1) * 16);

#pragma unroll
    for (int kk = 0; kk < 16; kk += 4) {
      const int ar = wv * 16 + l15;
      v2f aR = *(const v2f*)&sAr[cur][ar][kk + koff];
      v2f aI = *(const v2f*)&sAi[cur][ar][kk + koff];
      v2f br[4], bi[4];
#pragma unroll
      for (int s = 0; s < 4; ++s) {
        br[s] = *(const v2f*)&sBr[cur][s * 16 + l15][kk + koff];
        bi[s] = *(const v2f*)&sBi[cur][s * 16 + l15][kk + koff];
      }
#pragma unroll
      for (int s = 0; s < 4; ++s) {
        fP[s] = wmma_f32x4(aR, br[s], fP[s]);
        fQ[s] = wmma_f32x4(aI, bi[s], fQ[s]);
        fI[s] = wmma_f32x4(aR, bi[s], fI[s]);
        fI[s] = wmma_f32x4(aI, br[s], fI[s]);
      }
    }

    __builtin_amdgcn_s_wait_tensorcnt(0);
    __syncthreads();
  }

  // power spectrum + fftshift-aware radial binning into LDS histogram
  const int ub = u0 + wv * 16 + ((lane >> 4) << 3);
#pragma unroll
  for (int s = 0; s < 4; ++s) {
    const int v = v0 + s * 16 + l15;
    const int q = (v + 256) & 511;  // fftshift (self-inverse for N/2 shift)
#pragma unroll
    for (int g = 0; g < 8; ++g) {
      const int u = ub + g;
      const int p = (u + 256) & 511;
      const float fr = fP[s][g] - fQ[s][g];
      const float fi = fI[s][g];
      atomicAdd(&lbins[rbin[p * NN + q]], fr * fr + fi * fi);  // ds_add_f32
    }
  }
  __syncthreads();
  for (int b = tid; b < NBINS; b += 128) {
    const float vsum = lbins[b];
    if (vsum != 0.0f) atomicAdd(&binsum[(size_t)batch * NBINS + b], vsum);
  }
}

// ---------------------------------------------------------------- finalize
__global__ void k_final(const float* __restrict__ binsum,
                        const float* __restrict__ counts,
                        float* __restrict__ out) {
  int i = blockIdx.x * blockDim.x + threadIdx.x;
  if (i >= 32 * NB) return;
  int b = i / NB;
  int r = (i % NB) + 1;  // output covers bins 1..255
  float c = counts[r];
  if (c == 0.0f) c = 1.0f;
  float val = binsum[(size_t)b * NBINS + r] / (3.0f * c);  // channel mean /3
  out[i] = logf(val + 1e-8f);
}

// ------------------------------------------------------------------ launch
extern "C" void kernel_launch(void* const* d_in, const int* in_sizes, int n_in,
                              void* d_out, int out_size, void* d_ws,
                              size_t ws_size, hipStream_t stream) {
  const float* x = (const float*)d_in[0];    // [32,3,512,512] f32
  const int* rbin = (const int*)d_in[1];     // [512*512] i32 radial bins
  float* out = (float*)d_out;                // [32,255] f32

  float* ws = (float*)d_ws;
  float* Wr = ws;                            // 1 MB
  float* Wi = Wr + IMG;                      // 1 MB
  float* binsum = Wi + IMG;                  // 32*363 f32
  float* counts = binsum + 32 * NBINS;       // 363 f32 (padded to 512)
  float* T = counts + 512;                   // group intermediate: 2 MB/image

  const size_t fixedBytes = ((size_t)2 * IMG + 32 * NBINS + 512) * sizeof(float);
  const size_t perImg = (size_t)2 * IMG * sizeof(float);
  size_t avail = (ws_size > fixedBytes) ? (ws_size - fixedBytes) : 0;
  int G = (int)(avail / perImg);
  if (G < 1) G = 1;
  if (G > NIMG) G = NIMG;

  k_build_w<<<IMG / 256, 256, 0, stream>>>(Wr, Wi);
  k_zero<<<(32 * NBINS + 512 + 255) / 256, 256, 0, stream>>>(binsum,
                                                             32 * NBINS + 512);
  k_count<<<IMG / 256, 256, 0, stream>>>(rbin, counts);

  for (int g0 = 0; g0 < NIMG; g0 += G) {
    const int gc = (NIMG - g0 < G) ? (NIMG - g0) : G;
    k_fft_rows<<<dim3(8, 8, gc), 128, 0, stream>>>(x, Wr, Wi, T, g0);
    k_fft_cols_bin<<<dim3(8, 8, gc), 128, 0, stream>>>(Wr, Wi, T, rbin, binsum,
                                                       g0);
  }

  k_final<<<(32 * NB + 255) / 256, 256, 0, stream>>>(binsum, counts, out);
}